// MultiPosCrossEntropyLoss_28527172780416
// MI455X (gfx1250) — compile-verified
//
#include <hip/hip_runtime.h>
#include <hip/hip_bf16.h>
#include <math.h>

#define THREADS 256

constexpr int   C_FIXED = 4096;
constexpr int   CHUNK   = THREADS * 4;        // 1024 elements per staged chunk
constexpr int   NCHUNK  = C_FIXED / CHUNK;    // 4 chunks per row
constexpr float NEG_BIG = -1.0e30f;           // finite "-inf" sentinel (NaN-free masking)
constexpr float LOG2E   = 1.4426950408889634f;
constexpr float LN2     = 0.6931471805599453f;

template <int N>
__device__ __forceinline__ void wait_asynccnt() {
    asm volatile("s_wait_asynccnt %0" :: "i"(N) : "memory");
}

// ---- kernel 1: per-row weighted loss, one workgroup (8 wave32) per row --

__global__ __launch_bounds__(THREADS) void mpce_row_kernel(
    const float* __restrict__ pred, const int* __restrict__ label,
    const float* __restrict__ weight, float* __restrict__ row_out) {
    __shared__ __align__(16) float s_pred[C_FIXED];
    __shared__ __align__(16) int   s_lab[C_FIXED];
    __shared__ float w_mn[THREADS / 32], w_mp[THREADS / 32];
    __shared__ float w_sn[THREADS / 32], w_sp[THREADS / 32];

    const int t   = threadIdx.x;
    const int row = blockIdx.x;
    const float* gp = pred  + (size_t)row * C_FIXED;
    const int*   gl = label + (size_t)row * C_FIXED;

    // ---- stage the whole row into LDS via CDNA5 async DMA (8 ops/lane) --
#pragma unroll
    for (int c = 0; c < NCHUNK; ++c) {
        const int e = c * CHUNK + t * 4;
        unsigned long long ga = (unsigned long long)(uintptr_t)(gp + e);
        unsigned int       la = (unsigned int)(uintptr_t)(&s_pred[e]);
        asm volatile("global_load_async_to_lds_b128 %0, %1, off"
                     :: "v"(la), "v"(ga) : "memory");
        unsigned long long gb = (unsigned long long)(uintptr_t)(gl + e);
        unsigned int       lb = (unsigned int)(uintptr_t)(&s_lab[e]);
        asm volatile("global_load_async_to_lds_b128 %0, %1, off"
                     :: "v"(lb), "v"(gb) : "memory");
    }

    // ---- pass 1: masked per-class maxes, pipelined against the DMA ------
    float Mn = NEG_BIG;   // max over label==0 of  pred
    float Mp = NEG_BIG;   // max over label==1 of -pred

    auto chunk_max = [&](int c) {
        const int e = c * CHUNK + t * 4;
        float4 p = *reinterpret_cast<const float4*>(&s_pred[e]);
        int4   l = *reinterpret_cast<const int4*>(&s_lab[e]);
        Mn = fmaxf(Mn, (l.x == 0) ? p.x : NEG_BIG);
        Mp = fmaxf(Mp, (l.x == 0) ? NEG_BIG : -p.x);
        Mn = fmaxf(Mn, (l.y == 0) ? p.y : NEG_BIG);
        Mp = fmaxf(Mp, (l.y == 0) ? NEG_BIG : -p.y);
        Mn = fmaxf(Mn, (l.z == 0) ? p.z : NEG_BIG);
        Mp = fmaxf(Mp, (l.z == 0) ? NEG_BIG : -p.z);
        Mn = fmaxf(Mn, (l.w == 0) ? p.w : NEG_BIG);
        Mp = fmaxf(Mp, (l.w == 0) ? NEG_BIG : -p.w);
    };
    wait_asynccnt<6>(); chunk_max(0);   // async loads retire in order:
    wait_asynccnt<4>(); chunk_max(1);   // cnt<=6-2c  =>  chunk c is in LDS
    wait_asynccnt<2>(); chunk_max(2);
    wait_asynccnt<0>(); chunk_max(3);

    // wave32 butterfly max, then block max via LDS
    for (int off = 16; off > 0; off >>= 1) {
        Mn = fmaxf(Mn, __shfl_xor(Mn, off, 32));
        Mp = fmaxf(Mp, __shfl_xor(Mp, off, 32));
    }
    const int lane = t & 31, wave = t >> 5;
    if (lane == 0) { w_mn[wave] = Mn; w_mp[wave] = Mp; }
    __syncthreads();
#pragma unroll
    for (int w = 0; w < THREADS / 32; ++w) {
        Mn = fmaxf(Mn, w_mn[w]);
        Mp = fmaxf(Mp, w_mp[w]);
    }

    // ---- pass 2: Sum exp2(fma(v, log2e, -M*log2e)); masked terms -> 0 ---
    // Empty-class guard: if no element of a class exists, force the bias to
    // -inf so every term is exp2(-inf)=0 -> S=0 -> lse=-inf -> loss=0,
    // matching the reference's logaddexp(0, -inf) = 0.
    const float nMn2 = (Mn == NEG_BIG) ? -__builtin_inff() : -Mn * LOG2E;
    const float nMp2 = (Mp == NEG_BIG) ? -__builtin_inff() : -Mp * LOG2E;
    float Sn = 0.0f, Sp = 0.0f;
#pragma unroll
    for (int c = 0; c < NCHUNK; ++c) {
        const int e = c * CHUNK + t * 4;
        float4 p = *reinterpret_cast<const float4*>(&s_pred[e]);
        int4   l = *reinterpret_cast<const int4*>(&s_lab[e]);
        float vnx = (l.x == 0) ? p.x : NEG_BIG, vpx = (l.x == 0) ? NEG_BIG : -p.x;
        float vny = (l.y == 0) ? p.y : NEG_BIG, vpy = (l.y == 0) ? NEG_BIG : -p.y;
        float vnz = (l.z == 0) ? p.z : NEG_BIG, vpz = (l.z == 0) ? NEG_BIG : -p.z;
        float vnw = (l.w == 0) ? p.w : NEG_BIG, vpw = (l.w == 0) ? NEG_BIG : -p.w;
        Sn += exp2f(__fmaf_rn(vnx, LOG2E, nMn2));
        Sp += exp2f(__fmaf_rn(vpx, LOG2E, nMp2));
        Sn += exp2f(__fmaf_rn(vny, LOG2E, nMn2));
        Sp += exp2f(__fmaf_rn(vpy, LOG2E, nMp2));
        Sn += exp2f(__fmaf_rn(vnz, LOG2E, nMn2));
        Sp += exp2f(__fmaf_rn(vpz, LOG2E, nMp2));
        Sn += exp2f(__fmaf_rn(vnw, LOG2E, nMn2));
        Sp += exp2f(__fmaf_rn(vpw, LOG2E, nMp2));
    }

    // wave32 butterfly sum, then block sum + finalize on thread 0
    for (int off = 16; off > 0; off >>= 1) {
        Sn += __shfl_xor(Sn, off, 32);
        Sp += __shfl_xor(Sp, off, 32);
    }
    if (lane == 0) { w_sn[wave] = Sn; w_sp[wave] = Sp; }
    __syncthreads();

    if (t == 0) {
        float sn = 0.0f, sp = 0.0f;
#pragma unroll
        for (int w = 0; w < THREADS / 32; ++w) { sn += w_sn[w]; sp += w_sp[w]; }
        float lse_neg = Mn + log2f(sn) * LN2;   // log2f(0) = -inf handles empty class
        float lse_pos = Mp + log2f(sp) * LN2;
        float z = lse_neg + lse_pos;
        // logaddexp(0, z): stable for any z (z = -inf -> 0)
        float loss = fmaxf(z, 0.0f) + log1pf(expf(-fabsf(z)));
        row_out[row] = loss * weight[row];
    }
}

// ---- kernel 2: deterministic mean over N row losses ---------------------

__global__ __launch_bounds__(THREADS) void mpce_mean_kernel(
    const float* __restrict__ row_out, float* __restrict__ out, int n) {
    __shared__ float sm[THREADS];
    float acc = 0.0f;
    for (int i = threadIdx.x; i < n; i += THREADS) acc += row_out[i];
    sm[threadIdx.x] = acc;
    __syncthreads();
    for (int s = THREADS / 2; s > 0; s >>= 1) {
        if (threadIdx.x < s) sm[threadIdx.x] += sm[threadIdx.x + s];
        __syncthreads();
    }
    if (threadIdx.x == 0) out[0] = sm[0] / (float)n;
}

extern "C" void kernel_launch(void* const* d_in, const int* in_sizes, int n_in,
                              void* d_out, int out_size, void* d_ws, size_t ws_size,
                              hipStream_t stream) {
    const float* pred   = (const float*)d_in[0];   // [N, 4096] f32
    const int*   label  = (const int*)d_in[1];     // [N, 4096] i32 (0/1)
    const float* weight = (const float*)d_in[2];   // [N]       f32

    const int n = in_sizes[2];          // 8192 rows
    float* row_loss = (float*)d_ws;     // n floats of scratch (fully rewritten each call)

    mpce_row_kernel<<<n, THREADS, 0, stream>>>(pred, label, weight, row_loss);
    mpce_mean_kernel<<<1, THREADS, 0, stream>>>(row_loss, (float*)d_out, n);
}